// GAT_VAE_2628519985964
// MI455X (gfx1250) — compile-verified
//
#include <hip/hip_runtime.h>
#include <math.h>

// ---------------------------------------------------------------------------
// GAT (2 layers) + VAE for MI455X / gfx1250.
// Dense GEMMs use V_WMMA_F32_16X16X4_F32 (full fp32 matrix core path) with
// compile-time K/NC so all fragment loads use immediate offsets.
// Sparse edge work is L2-resident gather/scatter with float atomics.
// ---------------------------------------------------------------------------

typedef float v2f __attribute__((ext_vector_type(2)));
typedef float v8f __attribute__((ext_vector_type(8)));

#define NEG_SLOPE 0.2f

// CAS-based float atomic max (portable, handles negative values; -inf init).
__device__ __forceinline__ void atomicMaxF(float* addr, float val) {
  unsigned int* ua = (unsigned int*)addr;
  unsigned int old = __float_as_uint(*addr);
  while (__uint_as_float(old) < val) {
    unsigned int assumed = old;
    old = atomicCAS(ua, assumed, __float_as_uint(val));
    if (old == assumed) break;
  }
}

// ---------------------------------------------------------------------------
// fp32 WMMA GEMM: C[M,NC] = act(A[M,K] @ B[K,NC] + bias)
// One wave (32 lanes) per 16x16 output tile. Requires M % 16 == 0 (true:
// N = 100000 = 6250 * 16), K % 4 == 0, NC % 16 == 0.
// V_WMMA_F32_16X16X4_F32 layouts (ISA 7.12.2, 32-bit):
//   A (16x4): lane m=lane&15 holds K = (lane>>4)*2 .. +1   -> float2
//   B (4x16): lane n=lane&15 holds K = (lane>>4)*2 .. +1   -> float2
//   C (16x16): vgpr r -> row r + (lane>>4)*8, col = lane&15
// K and NC are compile-time so every load/store gets an immediate offset.
// ---------------------------------------------------------------------------
template <int K, int NC, bool BIAS, bool RELU>
__global__ void wmma_gemm(const float* __restrict__ A, const float* __restrict__ B,
                          const float* __restrict__ bias, float* __restrict__ C) {
  const int lane = threadIdx.x;          // 0..31
  const int nIdx = lane & 15;
  const int kGrp = (lane >> 4) << 1;     // 0 or 2
  const int col  = blockIdx.x * 16 + nIdx;

  const float* Ap = A + ((size_t)blockIdx.y * 16 + nIdx) * K + kGrp;  // per-lane A base
  const float* Bp = B + (size_t)kGrp * NC + col;                      // per-lane B base

  v8f acc = {};
#pragma unroll
  for (int k = 0; k < K; k += 4) {
    v2f a = *(const v2f*)(Ap + k);       // 8B-aligned, immediate offset k*4
    v2f b;
    b.x = Bp[k * NC];                    // immediate offset k*NC*4
    b.y = Bp[k * NC + NC];               // immediate offset (k+1)*NC*4
    acc = __builtin_amdgcn_wmma_f32_16x16x4_f32(
        /*neg_a=*/false, a, /*neg_b=*/false, b,
        /*c_mod=*/(short)0, acc, /*reuse_a=*/false, /*reuse_b=*/false);
  }

  float bv = 0.0f;
  if (BIAS) bv = bias[col];
  float* Cp = C + ((size_t)blockIdx.y * 16 + (lane >> 4) * 8) * NC + col;
#pragma unroll
  for (int r = 0; r < 8; ++r) {
    float v = acc[r] + bv;
    if (RELU) v = fmaxf(v, 0.0f);
    Cp[r * NC] = v;                      // immediate offset r*NC*4
  }
}

// ---------------------------------------------------------------------------
// Per-(node,head) attention projections; also initializes segment max/denoms.
// h layout: [N, H*D] row-major, head-major within row.
// ---------------------------------------------------------------------------
template <int H, int D>
__global__ void alpha_kernel(const float* __restrict__ h,
                             const float* __restrict__ a_src,
                             const float* __restrict__ a_dst,
                             float* __restrict__ alpha_s, float* __restrict__ alpha_d,
                             float* __restrict__ m, float* __restrict__ denom,
                             int N_) {
  int idx = blockIdx.x * blockDim.x + threadIdx.x;  // n*H + hh
  if (idx >= N_ * H) return;
  int hh = idx & (H - 1);
  const float* hp = h + (size_t)idx * D;            // (n*H+hh)*D
  const float* as = a_src + hh * D;
  const float* ad = a_dst + hh * D;
  float ss = 0.0f, sd = 0.0f;
#pragma unroll 4
  for (int d = 0; d < D; ++d) {
    float v = hp[d];
    ss += v * as[d];
    sd += v * ad[d];
  }
  alpha_s[idx] = ss;
  alpha_d[idx] = sd;
  m[idx]     = -INFINITY;
  denom[idx] = 0.0f;
}

template <int H>
__global__ void edge_logit_max(const int* __restrict__ src, const int* __restrict__ dst,
                               const float* __restrict__ as_, const float* __restrict__ ad_,
                               float* __restrict__ m, int E_) {
  int idx = blockIdx.x * blockDim.x + threadIdx.x;
  if (idx >= E_ * H) return;
  int hh = idx & (H - 1);
  int e  = idx / H;
  int s = src[e], d = dst[e];
  float x = as_[s * H + hh] + ad_[d * H + hh];
  float ev = x > 0.0f ? x : NEG_SLOPE * x;
  atomicMaxF(&m[d * H + hh], ev);
}

template <int H>
__global__ void edge_expsum(const int* __restrict__ src, const int* __restrict__ dst,
                            const float* __restrict__ as_, const float* __restrict__ ad_,
                            const float* __restrict__ m, float* __restrict__ exbuf,
                            float* __restrict__ denom, int E_) {
  int idx = blockIdx.x * blockDim.x + threadIdx.x;
  if (idx >= E_ * H) return;
  int hh = idx & (H - 1);
  int e  = idx / H;
  int s = src[e], d = dst[e];
  float x = as_[s * H + hh] + ad_[d * H + hh];
  float ev = x > 0.0f ? x : NEG_SLOPE * x;
  float ex = __expf(ev - m[d * H + hh]);
  exbuf[idx] = ex;
  atomicAdd(&denom[d * H + hh], ex);
}

template <int H>
__global__ void edge_norm(const int* __restrict__ dst, const float* __restrict__ denom,
                          float* __restrict__ exbuf, int E_) {
  int idx = blockIdx.x * blockDim.x + threadIdx.x;
  if (idx >= E_ * H) return;
  int hh = idx & (H - 1);
  int e  = idx / H;
  exbuf[idx] = exbuf[idx] / (denom[dst[e] * H + hh] + 1e-16f);
}

// One thread per (edge, channel); indices/coefs broadcast within L1.
template <int H, int HD, int D>
__global__ void edge_aggregate(const int* __restrict__ src, const int* __restrict__ dst,
                               const float* __restrict__ hfeat, const float* __restrict__ coef,
                               float* __restrict__ agg, int E_) {
  int idx = blockIdx.x * blockDim.x + threadIdx.x;
  if (idx >= E_ * HD) return;
  int c = idx & (HD - 1);
  int e = idx / HD;                      // HD power of two -> shift
  int hh = c / D;
  int s = src[e], d = dst[e];
  float v = hfeat[(size_t)s * HD + c] * coef[e * H + hh];
  atomicAdd(&agg[(size_t)d * HD + c], v);
}

template <int C_>
__global__ void bias_act(float* __restrict__ p, const float* __restrict__ b,
                         int n, int relu) {
  int i = blockIdx.x * blockDim.x + threadIdx.x;
  if (i >= n) return;
  float v = p[i] + b[i & (C_ - 1)];
  if (relu) v = fmaxf(v, 0.0f);
  p[i] = v;
}

__global__ void fill_f32(float* __restrict__ p, float v, int n) {
  int i = blockIdx.x * blockDim.x + threadIdx.x;
  if (i < n) p[i] = v;
}

__global__ void vae_z(const float* __restrict__ mu, const float* __restrict__ logvar,
                      const float* __restrict__ eps, float* __restrict__ z, int n) {
  int i = blockIdx.x * blockDim.x + threadIdx.x;
  if (i >= n) return;
  float lv = logvar[i];
  lv = fminf(fmaxf(lv, -10.0f), 10.0f);
  z[i] = mu[i] + eps[i] * __expf(0.5f * lv);
}

// ---------------------------------------------------------------------------
extern "C" void kernel_launch(void* const* d_in, const int* in_sizes, int n_in,
                              void* d_out, int out_size, void* d_ws, size_t ws_size,
                              hipStream_t stream) {
  (void)n_in; (void)out_size; (void)ws_size;

  const float* x      = (const float*)d_in[0];
  const int*   ei     = (const int*)d_in[1];
  // d_in[2] = edge_attr (unused by reference: edge_dim=None)
  const float* eps    = (const float*)d_in[3];
  const float* W1     = (const float*)d_in[4];
  const float* a_src1 = (const float*)d_in[5];
  const float* a_dst1 = (const float*)d_in[6];
  const float* b1     = (const float*)d_in[7];
  const float* W2     = (const float*)d_in[8];
  const float* a_src2 = (const float*)d_in[9];
  const float* a_dst2 = (const float*)d_in[10];
  const float* b2     = (const float*)d_in[11];
  const float* fc1_w  = (const float*)d_in[12];
  const float* fc1_b  = (const float*)d_in[13];
  const float* mu_w   = (const float*)d_in[14];
  const float* mu_b   = (const float*)d_in[15];
  const float* lv_w   = (const float*)d_in[16];
  const float* lv_b   = (const float*)d_in[17];
  const float* fc3_w  = (const float*)d_in[18];
  const float* fc3_b  = (const float*)d_in[19];
  const float* fc4_w  = (const float*)d_in[20];
  const float* fc4_b  = (const float*)d_in[21];

  const int Nn = in_sizes[0] / 128;   // 100000
  const int Ee = in_sizes[1] / 2;     // 1600000
  const int* src = ei;
  const int* dst = ei + Ee;

  // Output layout: recon [N,64] | mu [N,16] | logvar [N,16] | gat_out [N,64]
  float* out    = (float*)d_out;
  float* recon  = out;
  float* mu     = out + (size_t)Nn * 64;
  float* logvar = out + (size_t)Nn * 80;
  float* gat    = out + (size_t)Nn * 96;

  // Workspace carve-up
  float* ws = (float*)d_ws;
  size_t o = 0;
  float* h1   = ws + o; o += (size_t)Nn * 128;  // GAT1 hidden; later VAE h1
  float* agg1 = ws + o; o += (size_t)Nn * 128;  // GAT1 agg / hrelu; later h3
  float* h2   = ws + o; o += (size_t)Nn * 64;   // GAT2 hidden
  float* as1  = ws + o; o += (size_t)Nn * 2;
  float* ad1  = ws + o; o += (size_t)Nn * 2;
  float* m1   = ws + o; o += (size_t)Nn * 2;
  float* dn1  = ws + o; o += (size_t)Nn * 2;
  float* as2  = ws + o; o += (size_t)Nn;
  float* ad2  = ws + o; o += (size_t)Nn;
  float* m2   = ws + o; o += (size_t)Nn;
  float* dn2  = ws + o; o += (size_t)Nn;
  float* zb   = ws + o; o += (size_t)Nn * 16;
  float* ex1  = ws + o; o += (size_t)Ee * 2;
  float* ex2  = ws + o; o += (size_t)Ee;

  const dim3 wave(32, 1, 1);
  const int rowT = Nn / 16;              // 6250 (N % 16 == 0)
  const int T = 256;
  const int n128 = Nn * 128, n64 = Nn * 64, n16 = Nn * 16;
  const int eh1 = Ee * 2, ec1 = Ee * 128, ec2 = Ee * 64;

  // ---- GAT layer 1 ----
  wmma_gemm<128, 128, false, false><<<dim3(8, rowT), wave, 0, stream>>>(x, W1, nullptr, h1);
  alpha_kernel<2, 64><<<(Nn * 2 + T - 1) / T, T, 0, stream>>>(h1, a_src1, a_dst1, as1, ad1, m1, dn1, Nn);
  edge_logit_max<2><<<(eh1 + T - 1) / T, T, 0, stream>>>(src, dst, as1, ad1, m1, Ee);
  edge_expsum<2><<<(eh1 + T - 1) / T, T, 0, stream>>>(src, dst, as1, ad1, m1, ex1, dn1, Ee);
  edge_norm<2><<<(eh1 + T - 1) / T, T, 0, stream>>>(dst, dn1, ex1, Ee);
  fill_f32<<<(n128 + T - 1) / T, T, 0, stream>>>(agg1, 0.0f, n128);
  edge_aggregate<2, 128, 64><<<(ec1 + T - 1) / T, T, 0, stream>>>(src, dst, h1, ex1, agg1, Ee);
  bias_act<128><<<(n128 + T - 1) / T, T, 0, stream>>>(agg1, b1, n128, 1);   // relu

  // ---- GAT layer 2 ----
  wmma_gemm<128, 64, false, false><<<dim3(4, rowT), wave, 0, stream>>>(agg1, W2, nullptr, h2);
  alpha_kernel<1, 64><<<(Nn + T - 1) / T, T, 0, stream>>>(h2, a_src2, a_dst2, as2, ad2, m2, dn2, Nn);
  edge_logit_max<1><<<(Ee + T - 1) / T, T, 0, stream>>>(src, dst, as2, ad2, m2, Ee);
  edge_expsum<1><<<(Ee + T - 1) / T, T, 0, stream>>>(src, dst, as2, ad2, m2, ex2, dn2, Ee);
  edge_norm<1><<<(Ee + T - 1) / T, T, 0, stream>>>(dst, dn2, ex2, Ee);
  fill_f32<<<(n64 + T - 1) / T, T, 0, stream>>>(gat, 0.0f, n64);
  edge_aggregate<1, 64, 64><<<(ec2 + T - 1) / T, T, 0, stream>>>(src, dst, h2, ex2, gat, Ee);
  bias_act<64><<<(n64 + T - 1) / T, T, 0, stream>>>(gat, b2, n64, 0);       // no relu

  // ---- VAE ----
  wmma_gemm<64, 128, true, true><<<dim3(8, rowT), wave, 0, stream>>>(gat, fc1_w, fc1_b, h1);
  wmma_gemm<128, 16, true, false><<<dim3(1, rowT), wave, 0, stream>>>(h1, mu_w, mu_b, mu);
  wmma_gemm<128, 16, true, false><<<dim3(1, rowT), wave, 0, stream>>>(h1, lv_w, lv_b, logvar);
  vae_z<<<(n16 + T - 1) / T, T, 0, stream>>>(mu, logvar, eps, zb, n16);
  wmma_gemm<16, 128, true, true><<<dim3(8, rowT), wave, 0, stream>>>(zb, fc3_w, fc3_b, agg1);
  wmma_gemm<128, 64, true, false><<<dim3(4, rowT), wave, 0, stream>>>(agg1, fc4_w, fc4_b, recon);
}